// MyFeat_71837622992943
// MI455X (gfx1250) — compile-verified
//
#include <hip/hip_runtime.h>
#include <hip/hip_bf16.h>
#include <math.h>
#include <stdint.h>

#define T_WORDS 4000
#define NCHARS  16000
#define D_WORD  300
#define D_CHARE 768
#define D_POS   100
#define HDIM    400
#define DIN1    868     // D_CHARE + D_POS
#define DIN2    800     // 2*HDIM
#define G4H     1600    // 4*HDIM
#define KC      64      // K-chunk staged in LDS per double-buffer slot

typedef __attribute__((ext_vector_type(2))) float v2f;
typedef __attribute__((ext_vector_type(8))) float v8f;

__device__ __forceinline__ void async_wait_all() {
    asm volatile("s_wait_asynccnt 0x0" ::: "memory");
}

__device__ __forceinline__ void async_copy_b128(uint32_t lds_off, const float* gp) {
    // GLOBAL_LOAD_ASYNC_TO_LDS_B128: VDST = LDS byte address, VADDR = 64-bit global
    asm volatile("global_load_async_to_lds_b128 %0, %1, off"
                 :: "v"(lds_off), "v"(gp) : "memory");
}

// ---------------------------------------------------------------------------
// 1) chars_mean[w] = mean of char_seq_embeds[1+start : 1+end]  (seg_ids sorted)
// ---------------------------------------------------------------------------
__global__ void seg_mean_kernel(const float* __restrict__ char_embeds,
                                const int* __restrict__ seg_ids,
                                float* __restrict__ chars_mean)
{
    int w = blockIdx.x;
    int lo = 0, hi = NCHARS;
    while (lo < hi) { int mid = (lo + hi) >> 1; if (seg_ids[mid] < w) lo = mid + 1; else hi = mid; }
    int start = lo;
    hi = NCHARS;
    while (lo < hi) { int mid = (lo + hi) >> 1; if (seg_ids[mid] < w + 1) lo = mid + 1; else hi = mid; }
    int end = lo;
    float inv = 1.0f / (float)(end - start);
    for (int col = threadIdx.x; col < D_CHARE; col += blockDim.x) {
        float s = 0.0f;
        for (int r = start; r < end; ++r)
            s += char_embeds[(size_t)(r + 1) * D_CHARE + col];   // skip row 0 ([1:-1])
        chars_mean[(size_t)w * D_CHARE + col] = s * inv;
    }
}

// ---------------------------------------------------------------------------
// 2) ef[t, 768:868] = pos_table[pos_seq[t]]
// ---------------------------------------------------------------------------
__global__ void pos_embed_kernel(const int* __restrict__ pos_seq,
                                 const float* __restrict__ pos_table,
                                 float* __restrict__ ef)
{
    int t = blockIdx.x, j = threadIdx.x;
    if (j < D_POS)
        ef[(size_t)t * DIN1 + D_CHARE + j] =
            pos_table[(size_t)pos_seq[t] * D_POS + j];
}

// ---------------------------------------------------------------------------
// 3) ef[t, 0:768] = tanh(word_table[word_seq[t]] @ Ww^T + Wb) + chars_mean[t]
//    Block = 128 threads (4 waves). Output tile 32m x 64n. Gathered word rows
//    staged to LDS with async b128 copies (double buffered), B from global.
// ---------------------------------------------------------------------------
__global__ __launch_bounds__(128) void wc_wmma_kernel(
    const int* __restrict__ word_seq, const float* __restrict__ word_table,
    const float* __restrict__ Ww, const float* __restrict__ Wb,
    const float* __restrict__ chars_mean, float* __restrict__ ef)
{
    __shared__ __align__(16) float atile[2][32 * KC];

    int tid  = threadIdx.x;
    int wave = tid >> 5;
    int lane = tid & 31;
    int half = lane >> 4;
    int l16  = lane & 15;
    int n_base = (blockIdx.x * 4 + wave) * 16;   // over 768
    int m_base = blockIdx.y * 32;                // over 4000

    uint32_t lbase[2] = { (uint32_t)(uintptr_t)&atile[0][0],
                          (uint32_t)(uintptr_t)&atile[1][0] };

    // stage rows [m_base, m_base+32), cols [k0, k0+len) of gathered word_e
    auto stage = [&](int k0, int len, int buf) {
        int nch = len >> 2;                      // b128 chunks per row
        int total = 32 * nch;
        for (int i = tid; i < total; i += 128) {
            int row = i / nch;
            int ch  = i - row * nch;
            int wrow = word_seq[m_base + row];
            const float* gp = word_table + (size_t)wrow * D_WORD + k0 + ch * 4;
            async_copy_b128(lbase[buf] + (uint32_t)((row * KC + ch * 4) * 4), gp);
        }
    };

    const float* bp = Ww + (size_t)(n_base + l16) * D_WORD + 2 * half;

    stage(0, KC, 0);
    async_wait_all();
    __syncthreads();

    v8f acc0 = {}, acc1 = {};
    int buf = 0;
    for (int k0 = 0; k0 < D_WORD; k0 += KC) {
        int len = min(KC, D_WORD - k0);
        int k0n = k0 + KC;
        if (k0n < D_WORD) stage(k0n, min(KC, D_WORD - k0n), buf ^ 1);
        const float* As = atile[buf];
        for (int kk = 0; kk < len; kk += 4) {
            v2f a0 = *(const v2f*)&As[l16 * KC + kk + 2 * half];
            v2f a1 = *(const v2f*)&As[(16 + l16) * KC + kk + 2 * half];
            v2f b  = *(const v2f*)(bp + k0 + kk);
            acc0 = __builtin_amdgcn_wmma_f32_16x16x4_f32(false, a0, false, b,
                                                         (short)0, acc0, false, false);
            acc1 = __builtin_amdgcn_wmma_f32_16x16x4_f32(false, a1, false, b,
                                                         (short)0, acc1, false, false);
        }
        async_wait_all();
        __syncthreads();
        buf ^= 1;
    }

    int n = n_base + l16;
    float bias = Wb[n];
#pragma unroll
    for (int r = 0; r < 8; ++r) {
        int m0 = m_base + r + 8 * half;
        int m1 = m0 + 16;
        ef[(size_t)m0 * DIN1 + n] = tanhf(acc0[r] + bias) + chars_mean[(size_t)m0 * D_CHARE + n];
        ef[(size_t)m1 * DIN1 + n] = tanhf(acc1[r] + bias) + chars_mean[(size_t)m1 * D_CHARE + n];
    }
}

// ---------------------------------------------------------------------------
// 4) Z[s, :] = X[row(s)] @ W^T + bias   (row(s) = s or T-1-s), W: [1600 x dk]
//    Block = 128 threads (4 waves), output tile 32m x 64n, A staged via
//    async-to-LDS double buffering, B streamed from L2.
// ---------------------------------------------------------------------------
__global__ __launch_bounds__(128) void proj_wmma_kernel(
    const float* __restrict__ X, int ldx, int dk,
    const float* __restrict__ W, const float* __restrict__ bias,
    float* __restrict__ Z, int reverseX)
{
    __shared__ __align__(16) float atile[2][32 * KC];

    int tid  = threadIdx.x;
    int wave = tid >> 5;
    int lane = tid & 31;
    int half = lane >> 4;
    int l16  = lane & 15;
    int n_base = blockIdx.x * 64 + wave * 16;    // over 1600
    int m_base = blockIdx.y * 32;                // over 4000

    uint32_t lbase[2] = { (uint32_t)(uintptr_t)&atile[0][0],
                          (uint32_t)(uintptr_t)&atile[1][0] };

    auto stage = [&](int k0, int len, int buf) {
        int nch = len >> 2;
        int total = 32 * nch;
        for (int i = tid; i < total; i += 128) {
            int row = i / nch;
            int ch  = i - row * nch;
            int m = m_base + row;
            int gr = reverseX ? (T_WORDS - 1 - m) : m;
            const float* gp = X + (size_t)gr * ldx + k0 + ch * 4;
            async_copy_b128(lbase[buf] + (uint32_t)((row * KC + ch * 4) * 4), gp);
        }
    };

    const float* bp = W + (size_t)(n_base + l16) * dk + 2 * half;

    stage(0, min(KC, dk), 0);
    async_wait_all();
    __syncthreads();

    v8f acc0 = {}, acc1 = {};
    int buf = 0;
    for (int k0 = 0; k0 < dk; k0 += KC) {
        int len = min(KC, dk - k0);
        int k0n = k0 + KC;
        if (k0n < dk) stage(k0n, min(KC, dk - k0n), buf ^ 1);
        const float* As = atile[buf];
        for (int kk = 0; kk < len; kk += 4) {
            v2f a0 = *(const v2f*)&As[l16 * KC + kk + 2 * half];
            v2f a1 = *(const v2f*)&As[(16 + l16) * KC + kk + 2 * half];
            v2f b  = *(const v2f*)(bp + k0 + kk);
            acc0 = __builtin_amdgcn_wmma_f32_16x16x4_f32(false, a0, false, b,
                                                         (short)0, acc0, false, false);
            acc1 = __builtin_amdgcn_wmma_f32_16x16x4_f32(false, a1, false, b,
                                                         (short)0, acc1, false, false);
        }
        async_wait_all();
        __syncthreads();
        buf ^= 1;
    }

    int n = n_base + l16;
    float bv = bias[n];
#pragma unroll
    for (int r = 0; r < 8; ++r) {
        int m0 = m_base + r + 8 * half;
        Z[(size_t)m0 * G4H + n]        = acc0[r] + bv;
        Z[(size_t)(m0 + 16) * G4H + n] = acc1[r] + bv;
    }
}

// ---------------------------------------------------------------------------
// 5) Persistent LSTM recurrence. gridDim.x == 2 (block 0 = fwd set, 1 = bwd).
//    800 threads: thread tid computes z[tid] and z[tid+800]; threads <400 own
//    (h,c) for unit tid. h lives in LDS, c in a register. Whh streams from L2.
// ---------------------------------------------------------------------------
__global__ __launch_bounds__(800) void lstm_rec_kernel(
    const float* __restrict__ Zf, const float* __restrict__ WhF,
    const float* __restrict__ h0f, const float* __restrict__ c0f,
    float* __restrict__ outF, int ldF, int coF, int revF,
    const float* __restrict__ Zb, const float* __restrict__ WhB,
    const float* __restrict__ h0b, const float* __restrict__ c0b,
    float* __restrict__ outB, int ldB, int coB, int revB)
{
    const float* Z  = blockIdx.x ? Zb  : Zf;
    const float* Wh = blockIdx.x ? WhB : WhF;
    const float* h0 = blockIdx.x ? h0b : h0f;
    const float* c0 = blockIdx.x ? c0b : c0f;
    float* out = blockIdx.x ? outB : outF;
    int ld  = blockIdx.x ? ldB  : ldF;
    int co  = blockIdx.x ? coB  : coF;
    int rev = blockIdx.x ? revB : revF;

    __shared__ __align__(16) float h_sh[HDIM];
    __shared__ float z_sh[G4H];
    int tid = threadIdx.x;
    if (tid < HDIM) h_sh[tid] = h0[tid];
    float c = (tid < HDIM) ? c0[tid] : 0.0f;
    __syncthreads();

    int j0 = tid, j1 = tid + 800;
    const float4* w0 = (const float4*)(Wh + (size_t)j0 * HDIM);
    const float4* w1 = (const float4*)(Wh + (size_t)j1 * HDIM);

    for (int t = 0; t < T_WORDS; ++t) {
        if (t + 1 < T_WORDS) {  // hide next step's Z-row latency (global_prefetch_b8)
            __builtin_prefetch(Z + (size_t)(t + 1) * G4H + j0, 0, 1);
            __builtin_prefetch(Z + (size_t)(t + 1) * G4H + j1, 0, 1);
        }
        const float* zr = Z + (size_t)t * G4H;
        float a0 = zr[j0];
        float a1 = zr[j1];
        const float4* h4 = (const float4*)h_sh;
#pragma unroll 4
        for (int k = 0; k < HDIM / 4; ++k) {
            float4 hv = h4[k], x0 = w0[k], x1 = w1[k];
            a0 = fmaf(hv.x, x0.x, fmaf(hv.y, x0.y, fmaf(hv.z, x0.z, fmaf(hv.w, x0.w, a0))));
            a1 = fmaf(hv.x, x1.x, fmaf(hv.y, x1.y, fmaf(hv.z, x1.z, fmaf(hv.w, x1.w, a1))));
        }
        z_sh[j0] = a0;
        z_sh[j1] = a1;
        __syncthreads();
        if (tid < HDIM) {
            float zi = z_sh[tid];
            float zf = z_sh[tid + HDIM];
            float zg = z_sh[tid + 2 * HDIM];
            float zo = z_sh[tid + 3 * HDIM];
            float ig = 1.0f / (1.0f + __expf(-zi));
            float fg = 1.0f / (1.0f + __expf(-zf));
            float og = 1.0f / (1.0f + __expf(-zo));
            c = fg * c + ig * tanhf(zg);
            float hh = og * tanhf(c);
            h_sh[tid] = hh;
            int orow = rev ? (T_WORDS - 1 - t) : t;
            out[(size_t)orow * ld + co + tid] = hh;
        }
        __syncthreads();
    }
}

// ---------------------------------------------------------------------------
extern "C" void kernel_launch(void* const* d_in, const int* in_sizes, int n_in,
                              void* d_out, int out_size, void* d_ws, size_t ws_size,
                              hipStream_t stream)
{
    (void)in_sizes; (void)n_in; (void)out_size; (void)ws_size;
    const int*   word_seq    = (const int*)d_in[0];
    const int*   pos_seq     = (const int*)d_in[1];
    const int*   seg_ids     = (const int*)d_in[2];
    const float* char_embeds = (const float*)d_in[3];
    const float* word_table  = (const float*)d_in[4];
    const float* pos_table   = (const float*)d_in[5];
    const float* Ww          = (const float*)d_in[6];
    const float* Wb          = (const float*)d_in[7];
    const float* Wih1f = (const float*)d_in[8];
    const float* Whh1f = (const float*)d_in[9];
    const float* b1f   = (const float*)d_in[10];
    const float* h01f  = (const float*)d_in[11];
    const float* c01f  = (const float*)d_in[12];
    const float* Wih1b = (const float*)d_in[13];
    const float* Whh1b = (const float*)d_in[14];
    const float* b1b   = (const float*)d_in[15];
    const float* h01b  = (const float*)d_in[16];
    const float* c01b  = (const float*)d_in[17];
    const float* Wih2f = (const float*)d_in[18];
    const float* Whh2f = (const float*)d_in[19];
    const float* b2f   = (const float*)d_in[20];
    const float* h02f  = (const float*)d_in[21];
    const float* c02f  = (const float*)d_in[22];
    const float* Wih2b = (const float*)d_in[23];
    const float* Whh2b = (const float*)d_in[24];
    const float* b2b   = (const float*)d_in[25];
    const float* h02b  = (const float*)d_in[26];
    const float* c02b  = (const float*)d_in[27];

    float* out = (float*)d_out;                       // f1b [4000x400] ++ f2b [4000x400]

    // workspace layout (~90.2 MB of f32)
    float* ws         = (float*)d_ws;
    float* chars_mean = ws;                                     // 4000*768
    float* ef         = chars_mean + (size_t)T_WORDS * D_CHARE; // 4000*868
    float* l1         = ef + (size_t)T_WORDS * DIN1;            // 4000*800
    float* Zf         = l1 + (size_t)T_WORDS * DIN2;            // 4000*1600
    float* Zb         = Zf + (size_t)T_WORDS * G4H;             // 4000*1600

    seg_mean_kernel<<<T_WORDS, 256, 0, stream>>>(char_embeds, seg_ids, chars_mean);
    pos_embed_kernel<<<T_WORDS, 128, 0, stream>>>(pos_seq, pos_table, ef);
    wc_wmma_kernel<<<dim3(D_CHARE / 64, T_WORDS / 32), 128, 0, stream>>>(
        word_seq, word_table, Ww, Wb, chars_mean, ef);

    dim3 pgrid(G4H / 64, T_WORDS / 32);
    proj_wmma_kernel<<<pgrid, 128, 0, stream>>>(ef, DIN1, DIN1, Wih1f, b1f, Zf, 0);
    proj_wmma_kernel<<<pgrid, 128, 0, stream>>>(ef, DIN1, DIN1, Wih1b, b1b, Zb, 1);

    lstm_rec_kernel<<<2, 800, 0, stream>>>(
        Zf, Whh1f, h01f, c01f, l1, DIN2, 0,    0,
        Zb, Whh1b, h01b, c01b, l1, DIN2, HDIM, 1);

    proj_wmma_kernel<<<pgrid, 128, 0, stream>>>(l1, DIN2, DIN2, Wih2f, b2f, Zf, 0);
    proj_wmma_kernel<<<pgrid, 128, 0, stream>>>(l1, DIN2, DIN2, Wih2b, b2b, Zb, 1);

    lstm_rec_kernel<<<2, 800, 0, stream>>>(
        Zf, Whh2f, h02f, c02f, out,                          HDIM, 0, 0,
        Zb, Whh2b, h02b, c02b, out + (size_t)T_WORDS * HDIM, HDIM, 0, 0);
}